// MidpointMLR_17119739642309
// MI455X (gfx1250) — compile-verified
//
#include <hip/hip_runtime.h>

typedef __attribute__((ext_vector_type(2))) float v2f;
typedef __attribute__((ext_vector_type(4))) float v4f;
typedef __attribute__((ext_vector_type(8))) float v8f;

#define ROWS_PER_BLOCK 128
#define XS_STRIDE 68   // 64 + 4 pad: banks (4m+c)&63 distinct across lanes, keeps b128 align
#define ZT_STRIDE 68
#define EPSF 1e-15f

// PoincareLinear (HNN++ unidirectional MLR + ball projection), fp32 exact via
// V_WMMA_F32_16X16X4_F32.  One block = 128 rows x 64 cols of output.
__global__ __launch_bounds__(256) void
poincare_mlr_f32wmma(const float* __restrict__ x,
                     const float* __restrict__ z,
                     const float* __restrict__ bias,
                     float* __restrict__ out)
{
    __shared__ float xs[ROWS_PER_BLOCK * XS_STRIDE];  // staged x tile (padded)
    __shared__ float zt[64 * ZT_STRIDE];              // z_unit transposed [n][k]
    __shared__ float cx2s[ROWS_PER_BLOCK];            // per-row ||x||^2
    __shared__ float zpart[4 * 64];                   // column-norm partials
    __shared__ float coshd[64];
    __shared__ float sinhd[64];
    __shared__ float two_zn[64];

    const int tid  = threadIdx.x;
    const int lane = tid & 31;
    const int wave = tid >> 5;
    const int ln   = lane & 15;   // WMMA: M (A) / N (B,C) index
    const int kh   = lane >> 4;   // WMMA: K-half (A,B) / M+8 half (C)

    const float* xg = x + (size_t)blockIdx.x * ROWS_PER_BLOCK * 64;

    // warm the x tile while we preprocess z (gfx1250 global_prefetch_b8)
    __builtin_prefetch(xg + tid * 32, 0, 0);

    // ---- phase 1: z column-norm partials (all 256 threads) ----------------
    const int n  = tid & 63;   // column
    const int kq = tid >> 6;   // quarter of K
    float zreg[16];
    float ss = 0.0f;
#pragma unroll
    for (int j = 0; j < 16; ++j) {
        float v = z[(16 * kq + j) * 64 + n];
        zreg[j] = v;
        ss += v * v;
    }
    zpart[kq * 64 + n] = ss;
    __syncthreads();

    // ---- phase 2: finalize norms + bias transcendentals; zero cx2 --------
    if (tid < 64) {
        float nrm = sqrtf(zpart[n] + zpart[64 + n] + zpart[128 + n] + zpart[192 + n]);
        nrm = fmaxf(nrm, EPSF);
        zpart[n]  = 1.0f / nrm;      // broadcast inverse norm (reuse buffer)
        two_zn[n] = 2.0f * nrm;
        float d  = 2.0f * bias[n];   // drcr with rc = 1
        float e  = __expf(d);
        float ei = 1.0f / e;
        coshd[n] = 0.5f * (e + ei);
        sinhd[n] = 0.5f * (e - ei);
    } else if (tid >= 128) {
        cx2s[tid - 128] = 0.0f;
    }
    __syncthreads();

    // ---- phase 3: write z_unit^T; stage x tile + row sum-squares ---------
    {
        float inv = zpart[n];
#pragma unroll
        for (int j = 0; j < 16; ++j)
            zt[n * ZT_STRIDE + 16 * kq + j] = zreg[j] * inv;
    }
#pragma unroll
    for (int i = 0; i < 8; ++i) {
        int f4  = tid + 256 * i;          // float4 index within tile (coalesced)
        v4f v   = ((const v4f*)xg)[f4];
        int row = f4 >> 4;                // 16 float4 per row
        int c4  = f4 & 15;
        *(v4f*)&xs[row * XS_STRIDE + c4 * 4] = v;
        atomicAdd(&cx2s[row], v.x * v.x + v.y * v.y + v.z * v.z + v.w * v.w);
    }
    __syncthreads();

    // ---- phase 4: 16x64 @ 64x64 per wave via V_WMMA_F32_16X16X4_F32 ------
    const int rb = wave * 16;             // this wave's row stripe in the tile
    v8f acc[4] = {};                      // 4 N-tiles of 16x16 fp32
    const float* aBase = &xs[(rb + ln) * XS_STRIDE + 2 * kh];
#pragma unroll
    for (int s = 0; s < 16; ++s) {        // K = 64 in steps of 4
        v2f a = *(const v2f*)(aBase + 4 * s);
#pragma unroll
        for (int t = 0; t < 4; ++t) {
            v2f b = *(const v2f*)&zt[(ln + 16 * t) * ZT_STRIDE + 4 * s + 2 * kh];
            acc[t] = __builtin_amdgcn_wmma_f32_16x16x4_f32(
                false, a, false, b, (short)0, acc[t], false, false);
        }
    }

    // ---- phase 5: epilogue ------------------------------------------------
    float cd[4], sd[4], tz[4];
#pragma unroll
    for (int t = 0; t < 4; ++t) {
        int nc = ln + 16 * t;
        cd[t] = coshd[nc];
        sd[t] = sinhd[nc];
        tz[t] = two_zn[nc];
    }
    const size_t grow0 = (size_t)blockIdx.x * ROWS_PER_BLOCK + rb + 8 * kh;
#pragma unroll
    for (int r = 0; r < 8; ++r) {         // C layout: m = r + 8*kh
        float cx2    = cx2s[rb + 8 * kh + r];
        float opc    = 1.0f + cx2;
        float invden = 1.0f / fmaxf(1.0f - cx2, EPSF);
        float yv[4];
        float ssq = 0.0f;
#pragma unroll
        for (int t = 0; t < 4; ++t) {
            float num = 2.0f * acc[t][r] * cd[t] - opc * sd[t];
            float u   = num * invden;
            float w   = u + sqrtf(u * u + 1.0f);     // exp(asinh(u)), always > 0
            float e1  = __expf(tz[t] * __logf(w));   // w^(2*z_norm)
            float y   = 0.5f * (e1 - 1.0f / e1);     // sinh(2*z_norm*asinh(u))
            yv[t] = y;
            ssq  += y * y;
        }
        // per-row ||y||^2: reduce across the 16-lane half (rows disjoint per half)
        ssq += __shfl_xor(ssq, 1);
        ssq += __shfl_xor(ssq, 2);
        ssq += __shfl_xor(ssq, 4);
        ssq += __shfl_xor(ssq, 8);
        float scale = 1.0f / (1.0f + sqrtf(1.0f + ssq));
        float* orow = out + (grow0 + r) * 64;
#pragma unroll
        for (int t = 0; t < 4; ++t)
            orow[ln + 16 * t] = yv[t] * scale;
    }
}

extern "C" void kernel_launch(void* const* d_in, const int* in_sizes, int n_in,
                              void* d_out, int out_size, void* d_ws, size_t ws_size,
                              hipStream_t stream) {
    (void)n_in; (void)out_size; (void)d_ws; (void)ws_size;
    const float* x    = (const float*)d_in[0];
    const float* z    = (const float*)d_in[1];
    const float* bias = (const float*)d_in[2];
    float* out        = (float*)d_out;

    const int rows   = in_sizes[0] / 64;           // B = 1048576
    const int blocks = rows / ROWS_PER_BLOCK;      // 8192
    poincare_mlr_f32wmma<<<blocks, 256, 0, stream>>>(x, z, bias, out);
}